// Net_49761491091456
// MI455X (gfx1250) — compile-verified
//
#include <hip/hip_runtime.h>
#include <math.h>

typedef float v2f __attribute__((ext_vector_type(2)));
typedef float v8f __attribute__((ext_vector_type(8)));

#define F 128   // feature / hidden width
#define CLS 40  // output classes
#define CPAD 48 // padded class width (3 x 16 WMMA tiles)

// ---------------------------------------------------------------------------
// agg[i] = (1 + eps) * h[i]   (float4 vectorized)
// ---------------------------------------------------------------------------
__global__ __launch_bounds__(256) void init_agg(const float* __restrict__ H,
                                                float* __restrict__ Agg,
                                                const float* __restrict__ epsp,
                                                long long total4) {
  long long i = (long long)blockIdx.x * blockDim.x + threadIdx.x;
  if (i >= total4) return;
  float s = 1.0f + *epsp;
  float4 v = ((const float4*)H)[i];
  v.x *= s; v.y *= s; v.z *= s; v.w *= s;
  ((float4*)Agg)[i] = v;
}

// ---------------------------------------------------------------------------
// One wave per edge: lane l adds H[src][4l..4l+3] into Agg[dst][4l..4l+3].
// Atomics land in L2 (node matrix = 51 MB < 192 MB L2).
// ---------------------------------------------------------------------------
__global__ __launch_bounds__(256) void edge_scatter(const int* __restrict__ ei,
                                                    const float* __restrict__ H,
                                                    float* __restrict__ Agg,
                                                    long long E) {
  long long gid = (long long)blockIdx.x * blockDim.x + threadIdx.x;
  long long e = gid >> 5;
  int lane = threadIdx.x & 31;
  if (e >= E) return;
  long long src = (long long)ei[e];
  long long dst = (long long)ei[(size_t)E + e];
  float4 v = *(const float4*)(H + src * F + lane * 4);
  float* p = Agg + dst * F + lane * 4;
  atomicAdd(p + 0, v.x);
  atomicAdd(p + 1, v.y);
  atomicAdd(p + 2, v.z);
  atomicAdd(p + 3, v.w);
}

// ---------------------------------------------------------------------------
// C[N,128] = relu(A[N,128] @ W[128,128] + b)
// Block = 256 threads = 8 waves; block covers 128 rows, each wave one 16-row
// strip across all 8 column tiles. W cached in LDS; A fragments register-
// resident (v2f x 32) in the WMMA 16x4 f32 layout:
//   lanes 0-15 : M = lid, K = 4k+0..1 ; lanes 16-31 : M = lid, K = 4k+2..3
// ---------------------------------------------------------------------------
__global__ __launch_bounds__(256) void gemm_relu_128(const float* __restrict__ A,
                                                     const float* __restrict__ W,
                                                     const float* __restrict__ bias,
                                                     float* __restrict__ C,
                                                     int N) {
  __shared__ float Wlds[F * F];  // 64 KB
  for (int i = threadIdx.x; i < F * F / 4; i += 256)
    ((float4*)Wlds)[i] = ((const float4*)W)[i];
  __syncthreads();

  const int wave = threadIdx.x >> 5;
  const int lane = threadIdx.x & 31;
  const int half = lane >> 4;
  const int lid = lane & 15;

  const int row0 = blockIdx.x * 128 + wave * 16;
  int arow = row0 + lid;
  int arow_c = arow < N ? arow : (N - 1);  // clamp: keep EXEC all-ones for WMMA
  const float* Ap = A + (size_t)arow_c * F + 2 * half;

  v2f afrag[32];
#pragma unroll
  for (int kk = 0; kk < 32; ++kk) afrag[kk] = *(const v2f*)(Ap + kk * 4);

  for (int ct = 0; ct < 8; ++ct) {
    v8f acc = {};
    const float* Wp = Wlds + ct * 16 + lid + 2 * half * F;
#pragma unroll
    for (int kk = 0; kk < 32; ++kk) {
      v2f b;
      b.x = Wp[kk * 4 * F];
      b.y = Wp[kk * 4 * F + F];
      acc = __builtin_amdgcn_wmma_f32_16x16x4_f32(false, afrag[kk], false, b,
                                                  (short)0, acc, false, false);
    }
    float bv = bias[ct * 16 + lid];
#pragma unroll
    for (int r = 0; r < 8; ++r) {
      int row = row0 + r + 8 * half;  // D layout: half selects M=r vs M=8+r
      if (row < N) {
        float v = acc[r] + bv;
        C[(size_t)row * F + ct * 16 + lid] = v > 0.0f ? v : 0.0f;
      }
    }
  }
}

// ---------------------------------------------------------------------------
// Out[N,40] = log_softmax(A[N,128] @ W4[128,40] + b4)
// W4 zero-padded to 48 cols -> 3 WMMA col tiles. Tile spilled to per-wave LDS,
// then lanes 0..15 each reduce one 40-wide row.
// ---------------------------------------------------------------------------
__global__ __launch_bounds__(256) void final_logsoftmax(const float* __restrict__ A,
                                                        const float* __restrict__ W,
                                                        const float* __restrict__ bias,
                                                        float* __restrict__ Out,
                                                        int N) {
  __shared__ float Wlds[F * CPAD];      // 24 KB
  __shared__ float blds[CPAD];
  __shared__ float tile[8][16 * CPAD];  // 24 KB (per-wave scratch)

  for (int i = threadIdx.x; i < F * CPAD; i += 256) {
    int k = i / CPAD, n = i % CPAD;
    Wlds[i] = (n < CLS) ? W[k * CLS + n] : 0.0f;
  }
  if (threadIdx.x < CPAD)
    blds[threadIdx.x] = (threadIdx.x < CLS) ? bias[threadIdx.x] : 0.0f;
  __syncthreads();

  const int wave = threadIdx.x >> 5;
  const int lane = threadIdx.x & 31;
  const int half = lane >> 4;
  const int lid = lane & 15;

  const int row0 = blockIdx.x * 128 + wave * 16;
  int arow = row0 + lid;
  int arow_c = arow < N ? arow : (N - 1);
  const float* Ap = A + (size_t)arow_c * F + 2 * half;

  v2f afrag[32];
#pragma unroll
  for (int kk = 0; kk < 32; ++kk) afrag[kk] = *(const v2f*)(Ap + kk * 4);

  float* my = tile[wave];
  for (int ct = 0; ct < 3; ++ct) {
    v8f acc = {};
    const float* Wp = Wlds + ct * 16 + lid + 2 * half * CPAD;
#pragma unroll
    for (int kk = 0; kk < 32; ++kk) {
      v2f b;
      b.x = Wp[kk * 4 * CPAD];
      b.y = Wp[kk * 4 * CPAD + CPAD];
      acc = __builtin_amdgcn_wmma_f32_16x16x4_f32(false, afrag[kk], false, b,
                                                  (short)0, acc, false, false);
    }
    float bv = blds[ct * 16 + lid];
#pragma unroll
    for (int r = 0; r < 8; ++r)
      my[(r + 8 * half) * CPAD + ct * 16 + lid] = acc[r] + bv;
  }
  __syncthreads();

  if (lane < 16) {
    int row = row0 + lane;
    if (row < N) {
      const float* v = my + lane * CPAD;
      float mx = -INFINITY;
      for (int j = 0; j < CLS; ++j) mx = fmaxf(mx, v[j]);
      float s = 0.0f;
      for (int j = 0; j < CLS; ++j) s += expf(v[j] - mx);
      float lse = mx + logf(s);
      float* o = Out + (size_t)row * CLS;
      for (int j = 0; j < CLS; ++j) o[j] = v[j] - lse;
    }
  }
}

// ---------------------------------------------------------------------------
extern "C" void kernel_launch(void* const* d_in, const int* in_sizes, int n_in,
                              void* d_out, int out_size, void* d_ws, size_t ws_size,
                              hipStream_t stream) {
  const float* x   = (const float*)d_in[0];
  const int*  ei   = (const int*)d_in[1];   // edge_index, 2 x E
  const float* e1  = (const float*)d_in[2];
  const float* w1  = (const float*)d_in[3];
  const float* b1  = (const float*)d_in[4];
  const float* w2  = (const float*)d_in[5];
  const float* b2  = (const float*)d_in[6];
  const float* e2  = (const float*)d_in[7];
  const float* w3  = (const float*)d_in[8];
  const float* b3  = (const float*)d_in[9];
  const float* w4  = (const float*)d_in[10];
  const float* b4  = (const float*)d_in[11];
  float* out = (float*)d_out;

  const int N = in_sizes[0] / F;           // 100000
  const long long E = in_sizes[1] / 2;     // 1600000

  float* bufA = (float*)d_ws;
  float* bufB = bufA + (size_t)N * F;      // ping-pong buffers, 51.2 MB each

  const long long total4 = (long long)N * F / 4;
  const int gInit = (int)((total4 + 255) / 256);
  const int gEdge = (int)((E * 32 + 255) / 256);
  const int gGemm = (N + 127) / 128;
  dim3 blk(256);

  // conv1 aggregate: agg1 = (1+eps1)*x + scatter_sum(x)
  init_agg<<<gInit, blk, 0, stream>>>(x, bufA, e1, total4);
  edge_scatter<<<gEdge, blk, 0, stream>>>(ei, x, bufA, E);
  // MLP1: relu(.. @ w1 + b1), relu(.. @ w2 + b2)  (outer relu idempotent)
  gemm_relu_128<<<gGemm, blk, 0, stream>>>(bufA, w1, b1, bufB, N);
  gemm_relu_128<<<gGemm, blk, 0, stream>>>(bufB, w2, b2, bufA, N);
  // conv2 aggregate: agg2 = (1+eps2)*h + scatter_sum(h)
  init_agg<<<gInit, blk, 0, stream>>>(bufA, bufB, e2, total4);
  edge_scatter<<<gEdge, blk, 0, stream>>>(ei, bufA, bufB, E);
  // MLP2 first layer
  gemm_relu_128<<<gGemm, blk, 0, stream>>>(bufB, w3, b3, bufA, N);
  // final linear + log_softmax
  final_logsoftmax<<<gGemm, blk, 0, stream>>>(bufA, w4, b4, out, N);
}